// FeatureTokenizer_34342558499173
// MI455X (gfx1250) — compile-verified
//
#include <hip/hip_runtime.h>
#include <stdint.h>

#define NUM_FEATURES 256
#define EMBED_DIM    128
#define LN_EPS       1e-5f

#define FEATS         8    // features per block (== waves per block)
#define ROWS          32   // batch rows per block
#define ROWS_PER_WAVE 4    // 8 waves * 4 rows = 32 rows

typedef __attribute__((ext_vector_type(4))) float        f32x4;
typedef __attribute__((ext_vector_type(4))) unsigned int u32x4;
typedef __attribute__((ext_vector_type(4))) int          i32x4;
typedef __attribute__((ext_vector_type(8))) int          i32x8;

// ---------------------------------------------------------------------------
// Tensor Data Mover: DMA a 2D tile (dim1 rows x dim0 fp32 elements, row stride
// = dim0) from global memory into LDS at byte offset lds_off.
// D# layout per CDNA5 ISA. 6-arg builtin form.
// ---------------------------------------------------------------------------
__device__ __forceinline__ void tdm_load_2d_f32(const float* gptr,
                                                uint32_t lds_off,
                                                uint32_t dim0, uint32_t dim1) {
  const uint64_t ga = (uint64_t)(uintptr_t)gptr;
  const uint32_t s1 = dim0 * dim1;

  u32x4 g0;
  g0.x = 1u;                                   // count=1, is_restore=0, gather off
  g0.y = lds_off;                              // lds_addr (bytes)
  g0.z = (uint32_t)(ga & 0xFFFFFFFFu);         // global_addr[31:0]
  g0.w = (uint32_t)((ga >> 32) & 0x1FFFFFFu)   // global_addr[56:32]
       | (2u << 30);                           // type = 2 ("image")

  i32x8 g1;
  g1[0] = (int)(2u << 16);                                   // data_size=2 -> 4B
  g1[1] = (int)((dim0 & 0xFFFFu) << 16);                     // tensor_dim0[15:0]
  g1[2] = (int)((dim0 >> 16) | ((dim1 & 0xFFFFu) << 16));    // dim0 hi | dim1 lo
  g1[3] = (int)((dim1 >> 16) | ((dim0 & 0xFFFFu) << 16));    // dim1 hi | tile_dim0
  g1[4] = (int)(dim1 & 0xFFFFu);                             // tile_dim1 | tile_dim2=0
  g1[5] = (int)dim0;                                         // tensor_dim0_stride lo
  g1[6] = (int)((s1 & 0xFFFFu) << 16);                       // stride0 hi=0 | stride1 lo
  g1[7] = (int)(s1 >> 16);                                   // stride1 hi
  i32x4 g2 = {1, 0, 0, 0};                                   // tensor_dim2=1, no iterate
  i32x4 g3 = {0, 0, 0, 0};
  i32x8 g4 = {0, 0, 0, 0, 0, 0, 0, 0};                       // unused trailing group

  __builtin_amdgcn_tensor_load_to_lds(g0, g1, g2, g3, g4, 0);
}

__global__ __launch_bounds__(256) void FeatureTokenizer_kernel(
    const float* __restrict__ x,      // (B, F)
    const float* __restrict__ Wm,     // (F, D)
    const float* __restrict__ bm,     // (F, D)
    const float* __restrict__ Em,     // (F, D)
    const float* __restrict__ gamma,  // (D,)
    const float* __restrict__ beta,   // (D,)
    float* __restrict__ out) {        // (B, F, D)
  __shared__ __align__(16) float sW    [FEATS * EMBED_DIM]; // 4 KB
  __shared__ __align__(16) float sBE   [FEATS * EMBED_DIM]; // 4 KB (b, then b+E)
  __shared__ __align__(16) float sE    [FEATS * EMBED_DIM]; // 4 KB
  __shared__ __align__(16) float sStats[FEATS * 8];         // {a1,a0,q2,q1,q0'} per feat

  const int fbase = blockIdx.x * FEATS;
  const int bbase = blockIdx.y * ROWS;
  const int tid   = threadIdx.x;
  const int lane  = tid & 31;
  const int wave  = tid >> 5;
  const float inv = 1.0f / (float)EMBED_DIM;

  // ---- stage W / b / E feature rows via TDM (one wave issues; EXEC ignored) ----
  if (tid < 32) {
    if (lane == 0) {
      tdm_load_2d_f32(Wm + (size_t)fbase * EMBED_DIM,
                      (uint32_t)(uintptr_t)&sW[0], EMBED_DIM, FEATS);
      tdm_load_2d_f32(bm + (size_t)fbase * EMBED_DIM,
                      (uint32_t)(uintptr_t)&sBE[0], EMBED_DIM, FEATS);
      tdm_load_2d_f32(Em + (size_t)fbase * EMBED_DIM,
                      (uint32_t)(uintptr_t)&sE[0], EMBED_DIM, FEATS);
    }
    __builtin_amdgcn_s_wait_tensorcnt(0);
  }

  // per-lane gamma/beta + x gather (overlaps with DMA)
  const int d4 = lane * 4;
  const f32x4 gv = *(const f32x4*)&gamma[d4];
  const f32x4 bv = *(const f32x4*)&beta[d4];

  // lane l holds x[bbase + 4*wave + (l>>3)][fbase + (l&7)]
  const int xr = bbase + wave * ROWS_PER_WAVE + (lane >> 3);
  const int xc = fbase + (lane & 7);
  const float xall = x[(size_t)xr * NUM_FEATURES + xc];

  __syncthreads();

  // ---- fold b+E and compute per-feature LN stats (wave w owns feature w) ----
  // tid*4 == wave*EMBED_DIM + lane*4, so each wave folds exactly its feature row.
  {
    const f32x4 wv = *(const f32x4*)&sW[tid * 4];
    f32x4 be       = *(const f32x4*)&sBE[tid * 4];
    const f32x4 ee = *(const f32x4*)&sE[tid * 4];
    be.x += ee.x; be.y += ee.y; be.z += ee.z; be.w += ee.w;
    *(f32x4*)&sBE[tid * 4] = be;

    float pW   = (wv.x + wv.y) + (wv.z + wv.w);
    float pBE  = (be.x + be.y) + (be.z + be.w);
    float pW2  = fmaf(wv.x, wv.x, fmaf(wv.y, wv.y, fmaf(wv.z, wv.z, wv.w * wv.w)));
    float pBE2 = fmaf(be.x, be.x, fmaf(be.y, be.y, fmaf(be.z, be.z, be.w * be.w)));
    float pWBE = fmaf(wv.x, be.x, fmaf(wv.y, be.y, fmaf(wv.z, be.z, wv.w * be.w)));
#pragma unroll
    for (int off = 16; off >= 1; off >>= 1) {
      pW   += __shfl_xor(pW,   off, 32);
      pBE  += __shfl_xor(pBE,  off, 32);
      pW2  += __shfl_xor(pW2,  off, 32);
      pBE2 += __shfl_xor(pBE2, off, 32);
      pWBE += __shfl_xor(pWBE, off, 32);
    }
    if (lane == 0) {
      sStats[wave * 8 + 0] = pW * inv;                  // a1: mean = xv*a1 + a0
      sStats[wave * 8 + 1] = pBE * inv;                 // a0
      sStats[wave * 8 + 2] = pW2 * inv;                 // q2
      sStats[wave * 8 + 3] = 2.0f * pWBE * inv;         // q1
      sStats[wave * 8 + 4] = fmaf(pBE2, inv, LN_EPS);   // q0' = q0 + eps (folded)
    }
  }
  __syncthreads();

  // ---- store-bandwidth-bound main loop: 1 NT b128 store per token per lane ----
  const int row0 = bbase + wave * ROWS_PER_WAVE;

#pragma unroll
  for (int c = 0; c < FEATS; ++c) {
    const f32x4 wv  = *(const f32x4*)&sW [c * EMBED_DIM + d4];
    const f32x4 bev = *(const f32x4*)&sBE[c * EMBED_DIM + d4];
    const f32x4 st  = *(const f32x4*)&sStats[c * 8];   // {a1,a0,q2,q1}
    const float q0e = sStats[c * 8 + 4];
    const int f = fbase + c;
#pragma unroll
    for (int r = 0; r < ROWS_PER_WAVE; ++r) {
      const float xv = __shfl(xall, r * FEATS + c, 32);

      const float tx = fmaf(xv, wv.x, bev.x);
      const float ty = fmaf(xv, wv.y, bev.y);
      const float tz = fmaf(xv, wv.z, bev.z);
      const float tw = fmaf(xv, wv.w, bev.w);

      const float mean = fmaf(xv, st.x, st.y);                 // xv*a1 + a0
      const float m2e  = fmaf(xv, fmaf(xv, st.z, st.w), q0e);  // E[t^2] + eps
      const float vare = fmaf(-mean, mean, m2e);               // var + eps
      const float rstd = __builtin_amdgcn_rsqf(vare);          // v_rsq_f32
      const float mr   = -mean * rstd;

      f32x4 o;  // o = ((t - mean)*rstd) * gamma + beta
      o.x = fmaf(fmaf(tx, rstd, mr), gv.x, bv.x);
      o.y = fmaf(fmaf(ty, rstd, mr), gv.y, bv.y);
      o.z = fmaf(fmaf(tz, rstd, mr), gv.z, bv.z);
      o.w = fmaf(fmaf(tw, rstd, mr), gv.w, bv.w);

      f32x4* dst = (f32x4*)&out[((size_t)(row0 + r) * NUM_FEATURES + f) * EMBED_DIM + d4];
      __builtin_nontemporal_store(o, dst);  // stream 256 MB, don't rinse L2
    }
  }
}

extern "C" void kernel_launch(void* const* d_in, const int* in_sizes, int n_in,
                              void* d_out, int out_size, void* d_ws, size_t ws_size,
                              hipStream_t stream) {
  const float* x     = (const float*)d_in[0];
  const float* Wm    = (const float*)d_in[1];
  const float* bm    = (const float*)d_in[2];
  const float* Em    = (const float*)d_in[3];
  const float* gamma = (const float*)d_in[4];
  const float* beta  = (const float*)d_in[5];
  float* out = (float*)d_out;

  const int B = in_sizes[0] / NUM_FEATURES;  // 2048
  dim3 grid(NUM_FEATURES / FEATS, B / ROWS); // (32, 64)
  FeatureTokenizer_kernel<<<grid, 256, 0, stream>>>(x, Wm, bm, Em, gamma, beta, out);
}